// MANNNet_83958020702550
// MI455X (gfx1250) — compile-verified
//
#include <hip/hip_runtime.h>
#include <math.h>

// ---- problem dims (from reference) ----
#define B_       256
#define WAY_     4
#define SHOT_    5
#define SSUP     20      // WAY*SHOT
#define QN_      75
#define TSTEPS   95      // SSUP + QN
#define FEAT_    1600
#define CELL_    200
#define G_       800     // 4*CELL
#define RD_      160     // READS*MEM_D
#define CTRLIN   1960    // FEAT + CELL + RD
#define READS_   4
#define MEMD     40
#define SLOTS_   128
#define GAMMA_   0.95f
#define EPS_     1e-8f

// register tiling for the big input GEMM
#define MT 4   // 16-row M tiles per wave
#define NT 2   // 16-col N tiles per wave

typedef __attribute__((ext_vector_type(2))) float v2f;
typedef __attribute__((ext_vector_type(8))) float v8f;

// CDNA5 fp32 WMMA: D(16x16) = A(16x4) * B(4x16) + C, full wave32.
__device__ __forceinline__ v8f wmma4(v2f a, v2f b, v8f c) {
  return __builtin_amdgcn_wmma_f32_16x16x4_f32(false, a, false, b, (short)0, c,
                                               false, false);
}

__device__ __forceinline__ float sigmf(float x) { return 1.0f / (1.0f + expf(-x)); }

// ---------------------------------------------------------------------------
// zero-fill scratch state
__global__ void zero_kernel(float* p, long n) {
  long i = (long)blockIdx.x * blockDim.x + threadIdx.x;
  if (i < n) p[i] = 0.f;
}

// ---------------------------------------------------------------------------
// labgate[t][n] = emb_table[prev_label(t)] . W_ih[n, FEAT:FEAT+CELL] + b_ih[n] + b_hh[n]
__global__ void labgate_kernel(const float* __restrict__ emb,
                               const float* __restrict__ Wih,
                               const float* __restrict__ bih,
                               const float* __restrict__ bhh,
                               float* __restrict__ labg) {
  int idx = blockIdx.x * blockDim.x + threadIdx.x;
  if (idx >= TSTEPS * G_) return;
  int t = idx / G_, n = idx % G_;
  int lab = (t == 0) ? WAY_ : ((t <= SSUP) ? (t - 1) / SHOT_ : WAY_);
  const float* e = emb + (size_t)lab * CELL_;
  const float* w = Wih + (size_t)n * CTRLIN + FEAT_;
  float acc = bih[n] + bhh[n];
  for (int k = 0; k < CELL_; ++k) acc += e[k] * w[k];
  labg[idx] = acc;
}

// ---------------------------------------------------------------------------
// Big parallel GEMM (hoisted out of the recurrence):
//   gx[t][b][n] = x(t,b,:) . W_ih[n, 0:FEAT] + labgate[t][n]
// Register-tiled 4x2 WMMA tiles per wave: per k-step 6 loads feed 8 WMMAs
// (0.75 loads/WMMA) instead of 2 loads/WMMA.
__global__ void gemm_x_kernel(const float* __restrict__ sup,
                              const float* __restrict__ qry,
                              const float* __restrict__ Wih,
                              const float* __restrict__ labg,
                              float* __restrict__ gx) {
  const int lane = threadIdx.x;              // 32 threads = 1 wave
  const int n0 = blockIdx.x * (16 * NT);     // G/(16*NT) = 25 groups
  const int mt = blockIdx.y;                 // TSTEPS * (B/(16*MT)) = 95*4
  const int t  = mt >> 2;
  const int mb = (mt & 3) * (16 * MT);
  const int m  = lane & 15;
  const int kk = (lane >> 4) << 1;           // K sub-pair per lane half

  const float* A[MT];
#pragma unroll
  for (int i = 0; i < MT; ++i) {
    int brow = mb + i * 16 + m;
    A[i] = (t < SSUP)
        ? (sup + ((size_t)brow * SSUP + t) * FEAT_)
        : (qry + ((size_t)brow * QN_ + (t - SSUP)) * FEAT_);
  }
  const float* Bw[NT];
#pragma unroll
  for (int j = 0; j < NT; ++j)
    Bw[j] = Wih + (size_t)(n0 + j * 16 + m) * CTRLIN;   // B[k][n] = W_ih[n][k]

  v8f acc[MT][NT] = {};
  for (int k = 0; k < FEAT_; k += 4) {
    v2f a[MT], bf[NT];
#pragma unroll
    for (int i = 0; i < MT; ++i) a[i] = *(const v2f*)(A[i] + k + kk);
#pragma unroll
    for (int j = 0; j < NT; ++j) bf[j] = *(const v2f*)(Bw[j] + k + kk);
#pragma unroll
    for (int i = 0; i < MT; ++i)
#pragma unroll
      for (int j = 0; j < NT; ++j)
        acc[i][j] = wmma4(a[i], bf[j], acc[i][j]);
  }

  const int hi = lane >> 4;
  float* outb = gx + ((size_t)t * B_) * G_;
#pragma unroll
  for (int i = 0; i < MT; ++i) {
#pragma unroll
    for (int j = 0; j < NT; ++j) {
      int nout = n0 + j * 16 + (lane & 15);
      float lg = labg[t * G_ + nout];
#pragma unroll
      for (int v = 0; v < 8; ++v) {
        int mo = mb + i * 16 + v + (hi << 3);  // C/D layout: VGPRv -> M=v (+8 hi half)
        outb[(size_t)mo * G_ + nout] = acc[i][j][v] + lg;
      }
    }
  }
}

// ---------------------------------------------------------------------------
// Per-step recurrent GEMM (latency-bound: 1 tile/wave to spread across SIMDs):
//   gates[b][n] = gx_t[b][n] + h[b,:] . W_hh[n,:] + readout[b,:] . W_ih[n, 1800:1960]
__global__ void step_gemm_kernel(const float* __restrict__ gx_t,
                                 const float* __restrict__ h,
                                 const float* __restrict__ ro,
                                 const float* __restrict__ Whh,
                                 const float* __restrict__ Wih,
                                 float* __restrict__ gates) {
  const int lane = threadIdx.x;
  const int n0 = blockIdx.x * 16;         // 50
  const int mb = blockIdx.y * 16;         // 16
  const int m  = lane & 15;
  const int kk = (lane >> 4) << 1;
  const float* Ah = h  + (size_t)(mb + m) * CELL_;
  const float* Ar = ro + (size_t)(mb + m) * RD_;
  const float* Bh = Whh + (size_t)(n0 + m) * CELL_;
  const float* Br = Wih + (size_t)(n0 + m) * CTRLIN + (FEAT_ + CELL_);
  v8f acc = {};
  for (int k = 0; k < CELL_; k += 4)
    acc = wmma4(*(const v2f*)(Ah + k + kk), *(const v2f*)(Bh + k + kk), acc);
  for (int k = 0; k < RD_; k += 4)
    acc = wmma4(*(const v2f*)(Ar + k + kk), *(const v2f*)(Br + k + kk), acc);
  const int hi = lane >> 4;
  const int nout = n0 + (lane & 15);
#pragma unroll
  for (int v = 0; v < 8; ++v) {
    int mo = mb + v + (hi << 3);
    gates[(size_t)mo * G_ + nout] = acc[v] + gx_t[(size_t)mo * G_ + nout];
  }
}

// ---------------------------------------------------------------------------
// Fused per-step update: LSTM pointwise + projections + least-used write +
// memory update + cosine read + softmax + readout + logits. One block per
// batch element (state fully resident in LDS for the step).
__global__ void step_update_kernel(
    const float* __restrict__ gates, float* __restrict__ h, float* __restrict__ c,
    float* __restrict__ readout, float* __restrict__ read_w, float* __restrict__ used_w,
    float* __restrict__ memory,
    const float* __restrict__ Wk, const float* __restrict__ bk,
    const float* __restrict__ Wa, const float* __restrict__ ba,
    const float* __restrict__ Ws, const float* __restrict__ bs,
    const float* __restrict__ Wh, const float* __restrict__ bh,
    const float* __restrict__ Wr, const float* __restrict__ br,
    float* __restrict__ out, int t) {
  const int b = blockIdx.x;
  const int tid = threadIdx.x;  // 128
  __shared__ float sh_h[CELL_];
  __shared__ float sh_key[RD_];
  __shared__ float sh_add[RD_];
  __shared__ float sh_mem[SLOTS_][MEMD];
  __shared__ float sh_used[SLOTS_];
  __shared__ float sh_mask[SLOTS_];
  __shared__ float sh_wsum[SLOTS_];
  __shared__ float sh_minv[SLOTS_];
  __shared__ float sh_logit[READS_][SLOTS_];
  __shared__ float sh_rw[READS_][SLOTS_];
  __shared__ float sh_ro[RD_];
  __shared__ float sh_kinv[READS_];
  __shared__ float sh_rmax[READS_], sh_rsum[READS_];
  __shared__ float sh_sigma;
  __shared__ float sh_hlog[WAY_];

  const float* gb = gates + (size_t)b * G_;
  float* cb = c + (size_t)b * CELL_;
  float* hb = h + (size_t)b * CELL_;

  // LSTM pointwise: i,f,g,o split of gates
  for (int j = tid; j < CELL_; j += 128) {
    float ig = sigmf(gb[j]);
    float fg = sigmf(gb[CELL_ + j]);
    float gg = tanhf(gb[2 * CELL_ + j]);
    float og = sigmf(gb[3 * CELL_ + j]);
    float cn = fg * cb[j] + ig * gg;
    float hn = og * tanhf(cn);
    cb[j] = cn; hb[j] = hn; sh_h[j] = hn;
  }
  if (tid < SLOTS_) { sh_used[tid] = used_w[(size_t)b * SLOTS_ + tid]; sh_mask[tid] = 0.f; }
  __syncthreads();

  // projections from h_new: key(160) | add(160,tanh) | sigma(1) | h-logits(4)
  for (int o = tid; o < 2 * RD_ + 1 + WAY_; o += 128) {
    const float* wrow; float bias; int kind, idx;
    if (o < RD_)            { idx = o;             wrow = Wk + (size_t)idx * CELL_; bias = bk[idx]; kind = 0; }
    else if (o < 2 * RD_)   { idx = o - RD_;       wrow = Wa + (size_t)idx * CELL_; bias = ba[idx]; kind = 1; }
    else if (o == 2 * RD_)  { idx = 0;             wrow = Ws;                       bias = bs[0];   kind = 2; }
    else                    { idx = o - 2*RD_ - 1; wrow = Wh + (size_t)idx * CELL_; bias = bh[idx]; kind = 3; }
    float acc = bias;
    for (int k = 0; k < CELL_; ++k) acc += sh_h[k] * wrow[k];
    if (kind == 0)      sh_key[idx] = acc;
    else if (kind == 1) sh_add[idx] = tanhf(acc);
    else if (kind == 2) sh_sigma = sigmf(acc);
    else                sh_hlog[idx] = acc;
  }
  __syncthreads();

  // top_k(-used_w, READS): 4 smallest used_w, ties -> lowest index (stable)
  if (tid == 0) {
    for (int r = 0; r < READS_; ++r) {
      float best = 3.4e38f; int bi = 0;
      for (int n = 0; n < SLOTS_; ++n)
        if (sh_mask[n] == 0.f && sh_used[n] < best) { best = sh_used[n]; bi = n; }
      sh_mask[bi] = 1.f;
    }
  }
  __syncthreads();

  // write weights + memory erase/add + row norms (one slot per thread)
  const float sig = sh_sigma;
  if (tid < SLOTS_) {
    const int n = tid;
    float wv[READS_]; float wsum = 0.f;
#pragma unroll
    for (int r = 0; r < READS_; ++r) {
      float w = sig * read_w[((size_t)b * READS_ + r) * SLOTS_ + n] + (1.f - sig) * sh_mask[n];
      wv[r] = w; wsum += w;
    }
    sh_wsum[n] = wsum;
    float keep = 1.f - sh_mask[n];
    float* mrow = memory + ((size_t)b * SLOTS_ + n) * MEMD;
    float sq = 0.f;
    for (int d = 0; d < MEMD; ++d) {
      float mval = mrow[d] * keep;
#pragma unroll
      for (int r = 0; r < READS_; ++r) mval += wv[r] * sh_add[r * MEMD + d];
      mrow[d] = mval;
      sh_mem[n][d] = mval;
      sq += mval * mval;
    }
    sh_minv[n] = 1.f / fmaxf(sqrtf(sq), EPS_);
  }
  __syncthreads();

  if (tid < READS_) {
    float sq = 0.f;
    for (int d = 0; d < MEMD; ++d) { float kv = sh_key[tid * MEMD + d]; sq += kv * kv; }
    sh_kinv[tid] = 1.f / fmaxf(sqrtf(sq), EPS_);
  }
  __syncthreads();

  // cosine logits
  if (tid < SLOTS_) {
    const int n = tid;
#pragma unroll
    for (int r = 0; r < READS_; ++r) {
      float acc = 0.f;
      for (int d = 0; d < MEMD; ++d) acc += sh_key[r * MEMD + d] * sh_mem[n][d];
      sh_logit[r][n] = acc * sh_kinv[r] * sh_minv[n];
    }
  }
  __syncthreads();

  // softmax stats per read head
  if (tid < READS_) {
    float mx = -3.4e38f;
    for (int n = 0; n < SLOTS_; ++n) mx = fmaxf(mx, sh_logit[tid][n]);
    float s = 0.f;
    for (int n = 0; n < SLOTS_; ++n) s += expf(sh_logit[tid][n] - mx);
    sh_rmax[tid] = mx; sh_rsum[tid] = 1.f / s;
  }
  __syncthreads();

  // read weights + used_w update
  if (tid < SLOTS_) {
    const int n = tid;
    float rsum = 0.f;
#pragma unroll
    for (int r = 0; r < READS_; ++r) {
      float w = expf(sh_logit[r][n] - sh_rmax[r]) * sh_rsum[r];
      sh_rw[r][n] = w;
      read_w[((size_t)b * READS_ + r) * SLOTS_ + n] = w;
      rsum += w;
    }
    used_w[(size_t)b * SLOTS_ + n] = GAMMA_ * sh_used[n] + rsum + sh_wsum[n];
  }
  __syncthreads();

  // readout = read_w_new @ memory
  for (int o = tid; o < RD_; o += 128) {
    int r = o / MEMD, d = o % MEMD;
    float acc = 0.f;
    for (int n = 0; n < SLOTS_; ++n) acc += sh_rw[r][n] * sh_mem[n][d];
    sh_ro[o] = acc;
    readout[(size_t)b * RD_ + o] = acc;
  }
  __syncthreads();

  // final logits for query steps
  if (t >= SSUP && tid < WAY_) {
    float acc = sh_hlog[tid] + br[tid];
    const float* wr = Wr + (size_t)tid * RD_;
    for (int j = 0; j < RD_; ++j) acc += sh_ro[j] * wr[j];
    out[((size_t)b * QN_ + (t - SSUP)) * WAY_ + tid] = acc;
  }
}

// ---------------------------------------------------------------------------
extern "C" void kernel_launch(void* const* d_in, const int* in_sizes, int n_in,
                              void* d_out, int out_size, void* d_ws, size_t ws_size,
                              hipStream_t stream) {
  (void)in_sizes; (void)n_in; (void)out_size; (void)ws_size;
  const float* sup = (const float*)d_in[0];
  const float* qry = (const float*)d_in[1];
  const float* emb = (const float*)d_in[2];
  const float* Wih = (const float*)d_in[3];
  const float* Whh = (const float*)d_in[4];
  const float* bih = (const float*)d_in[5];
  const float* bhh = (const float*)d_in[6];
  const float* Wk  = (const float*)d_in[7];
  const float* bk  = (const float*)d_in[8];
  const float* Wa  = (const float*)d_in[9];
  const float* ba  = (const float*)d_in[10];
  const float* Ws  = (const float*)d_in[11];
  const float* bs  = (const float*)d_in[12];
  const float* Wh  = (const float*)d_in[13];
  const float* bh  = (const float*)d_in[14];
  const float* Wr  = (const float*)d_in[15];
  const float* br  = (const float*)d_in[16];
  float* out = (float*)d_out;

  float* ws = (float*)d_ws;
  size_t off = 0;
  float* gx   = ws + off; off += (size_t)TSTEPS * B_ * G_;
  float* labg = ws + off; off += (size_t)TSTEPS * G_;
  float* h    = ws + off; off += (size_t)B_ * CELL_;          // zeroed region start
  float* c    = ws + off; off += (size_t)B_ * CELL_;
  float* ro   = ws + off; off += (size_t)B_ * RD_;
  float* rw   = ws + off; off += (size_t)B_ * READS_ * SLOTS_;
  float* uw   = ws + off; off += (size_t)B_ * SLOTS_;
  float* mem  = ws + off; off += (size_t)B_ * SLOTS_ * MEMD;  // zeroed region end
  float* gates = ws + off; off += (size_t)B_ * G_;

  long nz = (long)B_ * (2 * CELL_ + RD_ + READS_ * SLOTS_ + SLOTS_ + SLOTS_ * MEMD);
  zero_kernel<<<(int)((nz + 255) / 256), 256, 0, stream>>>(h, nz);
  labgate_kernel<<<(TSTEPS * G_ + 255) / 256, 256, 0, stream>>>(emb, Wih, bih, bhh, labg);
  gemm_x_kernel<<<dim3(G_ / (16 * NT), TSTEPS * (B_ / (16 * MT))), 32, 0, stream>>>(
      sup, qry, Wih, labg, gx);

  for (int t = 0; t < TSTEPS; ++t) {
    step_gemm_kernel<<<dim3(G_ / 16, B_ / 16), 32, 0, stream>>>(
        gx + (size_t)t * B_ * G_, h, ro, Whh, Wih, gates);
    step_update_kernel<<<B_, 128, 0, stream>>>(gates, h, c, ro, rw, uw, mem,
        Wk, bk, Wa, ba, Ws, bs, Wh, bh, Wr, br, out, t);
  }
}